// GeneralNet_79216376808040
// MI455X (gfx1250) — compile-verified
//
#include <hip/hip_runtime.h>
#include <hip/hip_bf16.h>

typedef float v2f __attribute__((ext_vector_type(2)));
typedef float v8f __attribute__((ext_vector_type(8)));

#define NN 100000        // N_NODES (divisible by 16)
#define NE 3200000       // N_EDGES
#define NG 64            // N_GRAPHS
#define XIN 128
#define HID 32
#define EDIM 16

// ---------------------------------------------------------------------------
// Kernel 1: edge preprocessing (layer-invariant): EA_agg[dst] += edge_attr,
// deg[dst] += 1.  4 threads per edge, float4 per thread.
// ---------------------------------------------------------------------------
__global__ void edge_pre_kernel(const float4* __restrict__ ea4,
                                const int* __restrict__ eidx,
                                float* __restrict__ EA,
                                float* __restrict__ deg) {
    unsigned t = blockIdx.x * blockDim.x + threadIdx.x;
    if (t >= (unsigned)NE * 4u) return;
    int e = (int)(t >> 2);
    int c = (int)(t & 3u);
    int dst = eidx[NE + e];
    float4 v = ea4[(size_t)e * 4 + c];
    float* p = EA + (size_t)dst * EDIM + c * 4;
    atomicAdd(p + 0, v.x);
    atomicAdd(p + 1, v.y);
    atomicAdd(p + 2, v.z);
    atomicAdd(p + 3, v.w);
    if (c == 0) atomicAdd(deg + dst, 1.0f);
}

// ---------------------------------------------------------------------------
// Kernel 2: feature scatter-add  Xagg[dst, :] += h[src, :]
// One thread per (edge, 4-float chunk); shift = log2(din/4).
// ---------------------------------------------------------------------------
__global__ void scatter_kernel(const float4* __restrict__ h4,
                               const int* __restrict__ eidx,
                               float* __restrict__ Xagg,
                               unsigned work, int shift) {
    unsigned t = blockIdx.x * blockDim.x + threadIdx.x;
    if (t >= work) return;
    int e = (int)(t >> shift);
    int c = (int)(t & ((1u << shift) - 1u));
    int src = eidx[e];
    int dst = eidx[NE + e];
    int d4 = 1 << shift;
    float4 v = h4[(size_t)src * d4 + c];
    float* p = Xagg + ((size_t)dst * d4 + c) * 4;
    atomicAdd(p + 0, v.x);
    atomicAdd(p + 1, v.y);
    atomicAdd(p + 2, v.z);
    atomicAdd(p + 3, v.w);
}

// ---------------------------------------------------------------------------
// Kernel 3 (WMMA): fused per-node layer
//   out = ELU( Xagg@Wm + EA@We + H@Ws + deg*(bm+be) + bs )
// One wave computes one 16(node)x16(out-feature) f32 tile with
// V_WMMA_F32_16X16X4_F32, stepping K by 4 over all three GEMMs.
// ---------------------------------------------------------------------------
__device__ inline v8f wmma_k_loop(const float* __restrict__ Arow,
                                  const float* __restrict__ B,
                                  int K, int ldb, int coln, int kg, v8f acc) {
#pragma unroll 4
    for (int k0 = 0; k0 < K; k0 += 4) {
        int ka = k0 + 2 * kg;                 // A: K = 2*(lane/16) + vgpr
        v2f a = *(const v2f*)(Arow + ka);     // 8B-aligned (ka even, ldA even)
        v2f b;                                // B: K = 2*(lane/16) + vgpr, N = lane&15
        b.x = B[(size_t)ka * ldb + coln];
        b.y = B[(size_t)(ka + 1) * ldb + coln];
        acc = __builtin_amdgcn_wmma_f32_16x16x4_f32(
            /*neg_a=*/false, a, /*neg_b=*/false, b,
            /*c_mod=*/(short)0, acc, /*reuse_a=*/false, /*reuse_b=*/false);
    }
    return acc;
}

template <int DIN, int DOUT>
__global__ void fused_layer_kernel(const float* __restrict__ Xagg,
                                   const float* __restrict__ EA,
                                   const float* __restrict__ H,
                                   const float* __restrict__ deg,
                                   const float* __restrict__ Wm, const float* __restrict__ bm,
                                   const float* __restrict__ We, const float* __restrict__ be,
                                   const float* __restrict__ Ws, const float* __restrict__ bs,
                                   float* __restrict__ out) {
    constexpr int CT    = DOUT / 16;
    constexpr int TILES = (NN / 16) * CT;
    int wid = (int)((blockIdx.x * blockDim.x + threadIdx.x) >> 5);
    if (wid >= TILES) return;                  // wave-uniform guard (EXEC all-1s)
    int rt = wid / CT;
    int ct = wid - rt * CT;

    int lane = threadIdx.x & 31;
    int m    = lane & 15;
    int kg   = lane >> 4;                      // K-group (0/1)
    int row  = rt * 16 + m;                    // A-fragment row (node)
    int coln = ct * 16 + m;                    // B/C column (out feature), n = lane&15

    v8f acc = {0.f, 0.f, 0.f, 0.f, 0.f, 0.f, 0.f, 0.f};
    acc = wmma_k_loop(Xagg + (size_t)row * DIN, Wm, DIN,  DOUT, coln, kg, acc);
    acc = wmma_k_loop(EA   + (size_t)row * EDIM, We, EDIM, DOUT, coln, kg, acc);
    acc = wmma_k_loop(H    + (size_t)row * DIN, Ws, DIN,  DOUT, coln, kg, acc);

    float bmsg = bm[coln] + be[coln];
    float bsv  = bs[coln];
#pragma unroll
    for (int i = 0; i < 8; ++i) {              // C/D: M = i + 8*(lane/16)
        int r = rt * 16 + i + 8 * kg;
        float val = acc[i] + deg[r] * bmsg + bsv;
        val = (val > 0.f) ? val : expm1f(val); // ELU(alpha=1)
        out[(size_t)r * DOUT + coln] = val;
    }
}

// ---------------------------------------------------------------------------
// Kernel 4: pooled sums per graph (batch sorted -> run-length, few atomics)
// 32 lanes cover the 32 features of one node; each thread walks CHUNK nodes.
// ---------------------------------------------------------------------------
#define POOL_CHUNK 64
__global__ void pool_kernel(const float* __restrict__ h3,
                            const int* __restrict__ batch,
                            float* __restrict__ out,
                            float* __restrict__ counts) {
    int t   = blockIdx.x * blockDim.x + threadIdx.x;
    int col = t & 31;
    int ch  = t >> 5;
    int v0  = ch * POOL_CHUNK;
    if (v0 >= NN) return;
    int v1 = min(v0 + POOL_CHUNK, NN);
    int cur = batch[v0];
    float sum = 0.f, cnt = 0.f;
    for (int v = v0; v < v1; ++v) {
        int b = batch[v];
        if (b != cur) {
            atomicAdd(out + (size_t)cur * HID + col, sum);
            if (col == 0) atomicAdd(counts + cur, cnt);
            sum = 0.f; cnt = 0.f; cur = b;
        }
        sum += h3[(size_t)v * HID + col];
        cnt += 1.f;
    }
    atomicAdd(out + (size_t)cur * HID + col, sum);
    if (col == 0) atomicAdd(counts + cur, cnt);
}

__global__ void norm_kernel(float* __restrict__ out, const float* __restrict__ counts) {
    int t = blockIdx.x * blockDim.x + threadIdx.x;
    if (t >= NG * HID) return;
    out[t] /= fmaxf(counts[t / HID], 1.0f);
}

// ---------------------------------------------------------------------------
extern "C" void kernel_launch(void* const* d_in, const int* in_sizes, int n_in,
                              void* d_out, int out_size, void* d_ws, size_t ws_size,
                              hipStream_t stream) {
    const float* x     = (const float*)d_in[0];
    const float* ea    = (const float*)d_in[1];
    const int*   eidx  = (const int*)d_in[2];
    const int*   batch = (const int*)d_in[3];
    const float* Wm1 = (const float*)d_in[4];  const float* bm1 = (const float*)d_in[5];
    const float* We1 = (const float*)d_in[6];  const float* be1 = (const float*)d_in[7];
    const float* Ws1 = (const float*)d_in[8];  const float* bs1 = (const float*)d_in[9];
    const float* Wm2 = (const float*)d_in[10]; const float* bm2 = (const float*)d_in[11];
    const float* We2 = (const float*)d_in[12]; const float* be2 = (const float*)d_in[13];
    const float* Ws2 = (const float*)d_in[14]; const float* bs2 = (const float*)d_in[15];
    const float* Wm3 = (const float*)d_in[16]; const float* bm3 = (const float*)d_in[17];
    const float* We3 = (const float*)d_in[18]; const float* be3 = (const float*)d_in[19];
    const float* Ws3 = (const float*)d_in[20]; const float* bs3 = (const float*)d_in[21];

    float* ws    = (float*)d_ws;
    float* deg   = ws;                                   // [NN]
    float* EA    = deg  + (size_t)NN;                    // [NN*16]
    float* Xagg  = EA   + (size_t)NN * EDIM;             // [NN*128] (reused per layer)
    float* h1    = Xagg + (size_t)NN * XIN;              // [NN*32]  (later reused as h3)
    float* h2    = h1   + (size_t)NN * HID;              // [NN*64]
    float* cnts  = h2   + (size_t)NN * 2 * HID;          // [64]
    float* h3    = h1;                                   // h1 dead by layer 3
    float* outF  = (float*)d_out;

    // zero accumulators (memset nodes are graph-capturable)
    hipMemsetAsync(deg,  0, (size_t)NN * (1 + EDIM) * sizeof(float), stream);
    hipMemsetAsync(cnts, 0, NG * sizeof(float), stream);
    hipMemsetAsync(d_out, 0, (size_t)out_size * sizeof(float), stream);

    const int BLK = 256;
    // edge preprocessing (EA_agg, deg) — once, reused by all 3 layers
    {
        unsigned work = (unsigned)NE * 4u;
        edge_pre_kernel<<<(work + BLK - 1) / BLK, BLK, 0, stream>>>(
            (const float4*)ea, eidx, EA, deg);
    }

    auto gemm_blocks = [](int tiles) { return (tiles * 32 + 255) / 256; };

    // ---- layer 1: din=128, dout=32 ----
    hipMemsetAsync(Xagg, 0, (size_t)NN * XIN * sizeof(float), stream);
    {
        unsigned work = (unsigned)NE * (XIN / 4);
        scatter_kernel<<<(work + BLK - 1) / BLK, BLK, 0, stream>>>(
            (const float4*)x, eidx, Xagg, work, 5);
    }
    fused_layer_kernel<XIN, HID><<<gemm_blocks((NN / 16) * (HID / 16)), BLK, 0, stream>>>(
        Xagg, EA, x, deg, Wm1, bm1, We1, be1, Ws1, bs1, h1);

    // ---- layer 2: din=32, dout=64 ----
    hipMemsetAsync(Xagg, 0, (size_t)NN * HID * sizeof(float), stream);
    {
        unsigned work = (unsigned)NE * (HID / 4);
        scatter_kernel<<<(work + BLK - 1) / BLK, BLK, 0, stream>>>(
            (const float4*)h1, eidx, Xagg, work, 3);
    }
    fused_layer_kernel<HID, 2 * HID><<<gemm_blocks((NN / 16) * (2 * HID / 16)), BLK, 0, stream>>>(
        Xagg, EA, h1, deg, Wm2, bm2, We2, be2, Ws2, bs2, h2);

    // ---- layer 3: din=64, dout=32 ----
    hipMemsetAsync(Xagg, 0, (size_t)NN * 2 * HID * sizeof(float), stream);
    {
        unsigned work = (unsigned)NE * (2 * HID / 4);
        scatter_kernel<<<(work + BLK - 1) / BLK, BLK, 0, stream>>>(
            (const float4*)h2, eidx, Xagg, work, 4);
    }
    fused_layer_kernel<2 * HID, HID><<<gemm_blocks((NN / 16) * (HID / 16)), BLK, 0, stream>>>(
        Xagg, EA, h2, deg, Wm3, bm3, We3, be3, Ws3, bs3, h3);

    // ---- global mean pool ----
    {
        int chunks = (NN + POOL_CHUNK - 1) / POOL_CHUNK;
        int work = chunks * 32;
        pool_kernel<<<(work + BLK - 1) / BLK, BLK, 0, stream>>>(h3, batch, outF, cnts);
    }
    norm_kernel<<<(NG * HID + BLK - 1) / BLK, BLK, 0, stream>>>(outF, cnts);

    (void)in_sizes; (void)n_in; (void)ws_size;
}